// gate_17703855194728
// MI455X (gfx1250) — compile-verified
//
#include <hip/hip_runtime.h>

typedef float v2f __attribute__((ext_vector_type(2)));
typedef float v4f __attribute__((ext_vector_type(4)));
typedef float v8f __attribute__((ext_vector_type(8)));

#define BATCH 16
#define HH 224
#define WW 224
#define CC 64
#define HO 112
#define WO 112
#define NPATCH (HO * WO)   /* 12544 */
#define TOPK 1254

// ---------------------------------------------------------------------------
// Kernel 1: gating conv as WMMA f32 16x16x4.
// One wave produces 16 consecutive patch scores (same b, ho; wo = wo0..wo0+15).
// A (SRC0, 16x4) = gating weights, identical in every row (LDS broadcast).
// B (SRC1, 4x16) = patch data, column n = patch n.
// D[m][n] = sum_k w[k]*patch_n[k]; row 0 gives the 16 scores.
// A/B lane layout (ISA 7.12.2): lanes 0-15 hold K={4c,4c+1}, lanes 16-31 hold
// K={4c+2,4c+3}; for B the lane-low index is the column n.
// ---------------------------------------------------------------------------
__global__ __launch_bounds__(128) void conv_gate_wmma(
    const float* __restrict__ x, const float* __restrict__ w,
    float* __restrict__ g) {
  __shared__ __align__(16) float wlds[256];
  const int tid = threadIdx.x;
  // Stage weights once: flattened [kh][kw][c] order == in-patch memory order.
  wlds[2 * tid]     = w[2 * tid];
  wlds[2 * tid + 1] = w[2 * tid + 1];
  __syncthreads();

  const int wave = tid >> 5;
  const int lane = tid & 31;
  const int tile = blockIdx.x * 4 + wave;         // 16*112*7 = 12544 tiles
  const int b    = tile / (HO * (WO / 16));
  const int r    = tile % (HO * (WO / 16));
  const int ho   = r / (WO / 16);
  const int wo0  = (r % (WO / 16)) * 16;

  const float* rowA = x + (((size_t)b * HH + 2 * ho) * WW + 2 * wo0) * CC;
  const float* rowB = rowA + (size_t)WW * CC;     // next image row

  const int half = lane >> 4;                     // K sub-chunk selector
  const int n    = lane & 15;                     // patch (B column)
  const int koff = half * 2;

  v8f acc = {};
  // k = 0..127 live in rowA (kh=0), k = 128..255 in rowB (kh=1)
  #pragma unroll 8
  for (int c = 0; c < 32; ++c) {
    const int k = 4 * c + koff;
    v2f a = *(const v2f*)&wlds[k];
    v2f bv = *(const v2f*)(rowA + n * 128 + k);
    acc = __builtin_amdgcn_wmma_f32_16x16x4_f32(false, a, false, bv,
                                                (short)0, acc, false, false);
  }
  #pragma unroll 8
  for (int c = 0; c < 32; ++c) {
    const int k = 4 * c + koff;
    v2f a = *(const v2f*)&wlds[128 + k];
    v2f bv = *(const v2f*)(rowB + n * 128 + k);
    acc = __builtin_amdgcn_wmma_f32_16x16x4_f32(false, a, false, bv,
                                                (short)0, acc, false, false);
  }
  // D row M=0: acc[0] on lanes 0..15, N = lane.
  if (lane < 16) {
    g[(size_t)b * NPATCH + ho * WO + wo0 + lane] = acc[0];
  }
}

// ---------------------------------------------------------------------------
// Kernel 2: exact K-th largest per image via 4-pass radix select in LDS.
// Keys are order-preserving u32 transforms of the f32 scores.
// ---------------------------------------------------------------------------
__global__ __launch_bounds__(256) void topk_select(
    const float* __restrict__ g, float* __restrict__ thresh) {
  __shared__ unsigned keys[NPATCH];     // ~50 KB, fits 320 KB WGP LDS
  __shared__ unsigned hist[256];
  __shared__ unsigned s_prefix, s_remaining;

  const int b = blockIdx.x;
  const int tid = threadIdx.x;
  const float* gb = g + (size_t)b * NPATCH;

  for (int i = tid; i < NPATCH; i += 256) {
    unsigned bits = __float_as_uint(gb[i]);
    keys[i] = (bits & 0x80000000u) ? ~bits : (bits | 0x80000000u);
  }
  if (tid == 0) { s_prefix = 0u; s_remaining = TOPK; }
  __syncthreads();

  for (int pass = 3; pass >= 0; --pass) {
    hist[tid] = 0u;
    __syncthreads();
    const unsigned prefix = s_prefix;
    const int shift = pass * 8;
    for (int i = tid; i < NPATCH; i += 256) {
      const unsigned key = keys[i];
      if (pass == 3 || (key >> (shift + 8)) == prefix) {
        atomicAdd(&hist[(key >> shift) & 0xFFu], 1u);
      }
    }
    __syncthreads();
    if (tid == 0) {
      const unsigned rem = s_remaining;
      unsigned cum = 0;                 // keys strictly above bucket d
      for (int d = 255; d >= 0; --d) {
        const unsigned c = hist[d];
        if (cum + c >= rem) {
          s_prefix = (s_prefix << 8) | (unsigned)d;
          s_remaining = rem - cum;
          break;
        }
        cum += c;
      }
    }
    __syncthreads();
  }
  if (tid == 0) {
    const unsigned T = s_prefix;        // K-th largest key
    const unsigned bits = (T & 0x80000000u) ? (T ^ 0x80000000u) : ~T;
    thresh[b] = __uint_as_float(bits);
  }
}

// ---------------------------------------------------------------------------
// Kernel 3: out = x * (score >= thresh ? score : 0), 128-bit per thread.
// ---------------------------------------------------------------------------
__global__ __launch_bounds__(256) void apply_gate(
    const float* __restrict__ x, const float* __restrict__ g,
    const float* __restrict__ thresh, float* __restrict__ out) {
  const size_t i4 = (size_t)blockIdx.x * 256 + threadIdx.x;  // v4f index
  const size_t pix = i4 >> 4;                   // 16 v4f per pixel (C=64)
  const int wcol = (int)(pix % WW);
  const size_t t = pix / WW;
  const int h = (int)(t % HH);
  const int b = (int)(t / HH);

  const float s  = g[(size_t)b * NPATCH + (h >> 1) * WO + (wcol >> 1)];
  const float th = thresh[b];
  const float gate = (s >= th) ? s : 0.0f;

  v4f v = ((const v4f*)x)[i4];
  v *= gate;
  __builtin_nontemporal_store(v, &((v4f*)out)[i4]);
}

// ---------------------------------------------------------------------------
extern "C" void kernel_launch(void* const* d_in, const int* in_sizes, int n_in,
                              void* d_out, int out_size, void* d_ws,
                              size_t ws_size, hipStream_t stream) {
  const float* x = (const float*)d_in[0];        // [16,224,224,64]
  const float* w = (const float*)d_in[1];        // [2,2,64,1]
  float* out = (float*)d_out;                    // [16,224,224,64]

  float* g      = (float*)d_ws;                  // [16,12544] patch scores
  float* thresh = g + (size_t)BATCH * NPATCH;    // [16]

  // 1) gating conv (reads 205 MB)
  conv_gate_wmma<<<(BATCH * HO * (WO / 16)) / 4, 128, 0, stream>>>(x, w, g);
  // 2) per-image K-th-largest threshold (0.8 MB)
  topk_select<<<BATCH, 256, 0, stream>>>(g, thresh);
  // 3) gated multiply (reads 205 MB, writes 205 MB)
  const size_t n4 = (size_t)BATCH * HH * WW * CC / 4;   // 12,845,056
  apply_gate<<<(unsigned)(n4 / 256), 256, 0, stream>>>(x, g, thresh, out);
}